// GCNEncoder_13288628814617
// MI455X (gfx1250) — compile-verified
//
#include <hip/hip_runtime.h>

// GCN encoder: 3x (GEMM -> edge aggregate -> self+bias -> BN [-> ReLU])
// D = 64 features, fp32 end to end. GEMM uses V_WMMA_F32_16X16X4_F32.

#define FD 64
#define NLAYERS 3
#define BN_EPS 1e-5f
#define GROWS 32            // rows per GEMM block; 100000 % 32 == 0 (no tail)
#define LPAD 4              // LDS row pad: stride 68 floats = 272 B (16B-aligned,
                            // conflict-free fragment reads: bank = (4*row + k) % 64)

typedef __attribute__((ext_vector_type(2))) float v2f;
typedef __attribute__((ext_vector_type(8))) float v8f;

// ---------------------------------------------------------------------------
// H = X @ W   (X: [N x 64] row-major, W: [64 x 64] row-major)
// Block: 128 threads = 4 waves, 32 rows (two 16-row M-tiles per wave).
// Wave w handles columns [16w, 16w+16). K-loop over 64 in steps of 4:
// B fragment loaded once per k-step, reused for both M-tiles -> 32 WMMA/wave.
// A-frag (16x4 f32, ISA 7.12.2): lane%16 = M, lane/16 selects K pair.
// B-frag (4x16 f32): lane%16 = N, lane/16 selects K pair; W staged transposed.
// ---------------------------------------------------------------------------
__global__ __launch_bounds__(128)
void gcn_gemm_wmma(const float* __restrict__ X, const float* __restrict__ W,
                   float* __restrict__ H, int N)
{
    __shared__ float Xs[GROWS][FD + LPAD];   // 8.5 KB
    __shared__ float Wt[FD][FD + LPAD];      // 17 KB, Wt[c][k] = W[k][c]

    const int tid  = threadIdx.x;
    const int row0 = blockIdx.x * GROWS;

    // W (row-major [k][c]) -> transposed LDS
    for (int i = tid; i < FD * FD; i += 128) {
        int k = i >> 6, c = i & 63;
        Wt[c][k] = W[i];
    }
    // X tile: b128 global loads, expanded into padded LDS rows
    const float4* X4 = (const float4*)(X + (size_t)row0 * FD);
    for (int i = tid; i < GROWS * (FD / 4); i += 128) {
        int r = i >> 4, c4 = (i & 15) << 2;
        float4 v = X4[i];
        Xs[r][c4 + 0] = v.x;
        Xs[r][c4 + 1] = v.y;
        Xs[r][c4 + 2] = v.z;
        Xs[r][c4 + 3] = v.w;
    }
    __syncthreads();

    const int wave  = tid >> 5;
    const int lane  = tid & 31;
    const int l16   = lane & 15;
    const int khalf = (lane >> 4) * 2;   // 0 or 2
    const int c0    = wave * 16;

    v8f acc0 = {};
    v8f acc1 = {};
    #pragma unroll
    for (int k = 0; k < FD; k += 4) {
        v2f b, a0, a1;
        b.x  = Wt[c0 + l16][k + khalf];
        b.y  = Wt[c0 + l16][k + khalf + 1];
        a0.x = Xs[l16][k + khalf];
        a0.y = Xs[l16][k + khalf + 1];
        a1.x = Xs[16 + l16][k + khalf];
        a1.y = Xs[16 + l16][k + khalf + 1];
        // (neg_a, A, neg_b, B, c_mod, C, reuse_a, reuse_b)
        acc0 = __builtin_amdgcn_wmma_f32_16x16x4_f32(false, a0, false, b,
                                                     (short)0, acc0, false, false);
        acc1 = __builtin_amdgcn_wmma_f32_16x16x4_f32(false, a1, false, b,
                                                     (short)0, acc1, false, false);
    }

    // C/D layout: VGPR r -> M = r + 8*(lane/16); lane%16 -> N
    const int n     = c0 + l16;
    const int mbase = (lane >> 4) * 8;
    #pragma unroll
    for (int r = 0; r < 8; ++r) {
        H[(size_t)(row0 + mbase + r) * FD + n]      = acc0[r];
        H[(size_t)(row0 + 16 + mbase + r) * FD + n] = acc1[r];
    }
}

// ---------------------------------------------------------------------------
// Degree / dinv
// ---------------------------------------------------------------------------
__global__ void k_init_deg(float* __restrict__ deg, int N)
{
    int i = blockIdx.x * blockDim.x + threadIdx.x;
    if (i < N) deg[i] = 1.0f;   // self-loop
}

__global__ void k_count_deg(const int* __restrict__ dst, float* __restrict__ deg, int E)
{
    int e = blockIdx.x * blockDim.x + threadIdx.x;
    if (e < E) atomicAdd(&deg[dst[e]], 1.0f);
}

__global__ void k_make_dinv(const float* __restrict__ deg, float* __restrict__ dinv, int N)
{
    int i = blockIdx.x * blockDim.x + threadIdx.x;
    if (i < N) dinv[i] = rsqrtf(deg[i]);
}

__global__ void k_zero(float* __restrict__ p, size_t n)
{
    size_t i = (size_t)blockIdx.x * blockDim.x + threadIdx.x;
    if (i < n) p[i] = 0.0f;
}

// ---------------------------------------------------------------------------
// Z[dst] += H[src] * dinv[src]*dinv[dst]  (16 lanes per edge, 4 feats/lane)
// H and Z are 25.6 MB each -> resident in the 192 MB L2; random gather and
// f32 atomics stay on-chip.
// ---------------------------------------------------------------------------
__global__ __launch_bounds__(256)
void k_edge_agg(const float* __restrict__ H, const int* __restrict__ src,
                const int* __restrict__ dst, const float* __restrict__ dinv,
                float* __restrict__ Z, int E)
{
    int e = blockIdx.x * 16 + (threadIdx.x >> 4);
    if (e >= E) return;
    int s = src[e], d = dst[e];
    float norm = dinv[s] * dinv[d];
    int f = (threadIdx.x & 15) * 4;
    const float4 hv = *(const float4*)(H + (size_t)s * FD + f);
    float* zp = Z + (size_t)d * FD + f;
    atomicAdd(zp + 0, hv.x * norm);
    atomicAdd(zp + 1, hv.y * norm);
    atomicAdd(zp + 2, hv.z * norm);
    atomicAdd(zp + 3, hv.w * norm);
}

// ---------------------------------------------------------------------------
// Z += H * dinv^2 + b  (in place), and accumulate per-feature sum / sumsq.
// Block: 256 threads, 64 rows per block; thread t: feature t&63, row group t>>6.
// ---------------------------------------------------------------------------
__global__ __launch_bounds__(256)
void k_self_bias_stats(float* __restrict__ Z, const float* __restrict__ H,
                       const float* __restrict__ dinv, const float* __restrict__ b,
                       float* __restrict__ gsum, float* __restrict__ gsumsq, int N)
{
    __shared__ float s_sum[4][FD];
    __shared__ float s_sq[4][FD];
    const int t = threadIdx.x;
    const int f = t & 63;
    const int g = t >> 6;
    const int row0 = blockIdx.x * 64;
    const float bias = b[f];

    float sum = 0.0f, sq = 0.0f;
    for (int r = g; r < 64; r += 4) {
        int row = row0 + r;
        if (row < N) {
            float di = dinv[row];
            size_t idx = (size_t)row * FD + f;
            float z = Z[idx] + H[idx] * di * di + bias;
            Z[idx] = z;
            sum += z;
            sq  += z * z;
        }
    }
    s_sum[g][f] = sum;
    s_sq[g][f]  = sq;
    __syncthreads();
    if (t < FD) {
        float ts = s_sum[0][t] + s_sum[1][t] + s_sum[2][t] + s_sum[3][t];
        float tq = s_sq[0][t]  + s_sq[1][t]  + s_sq[2][t]  + s_sq[3][t];
        atomicAdd(&gsum[t],   ts);
        atomicAdd(&gsumsq[t], tq);
    }
}

// ---------------------------------------------------------------------------
// Y = (Z - mu) * rsqrt(var + eps) * gamma + beta  [+ ReLU]
// ---------------------------------------------------------------------------
__global__ __launch_bounds__(256)
void k_bn_norm(const float* __restrict__ Z, float* __restrict__ Y,
               const float* __restrict__ gsum, const float* __restrict__ gsumsq,
               const float* __restrict__ gamma, const float* __restrict__ beta,
               int N, int do_relu)
{
    size_t i = (size_t)blockIdx.x * blockDim.x + threadIdx.x;
    if (i >= (size_t)N * FD) return;
    int f = (int)(i & 63);
    float inv_n = 1.0f / (float)N;
    float mu  = gsum[f] * inv_n;
    float var = gsumsq[f] * inv_n - mu * mu;
    float y = (Z[i] - mu) * rsqrtf(var + BN_EPS) * gamma[f] + beta[f];
    if (do_relu) y = fmaxf(y, 0.0f);
    Y[i] = y;
}

// ---------------------------------------------------------------------------
static inline size_t align256(size_t x) { return (x + 255) & ~(size_t)255; }

extern "C" void kernel_launch(void* const* d_in, const int* in_sizes, int n_in,
                              void* d_out, int out_size, void* d_ws, size_t ws_size,
                              hipStream_t stream)
{
    const float* x      = (const float*)d_in[0];
    const int*   ei     = (const int*)d_in[1];
    const float* Ws     = (const float*)d_in[2];
    const float* bs     = (const float*)d_in[3];
    const float* gammas = (const float*)d_in[4];
    const float* betas  = (const float*)d_in[5];

    const int N = in_sizes[0] / FD;
    const int E = in_sizes[1] / 2;
    const int* src = ei;
    const int* dst = ei + E;

    const size_t nd   = (size_t)N * FD;
    const size_t nd_b = align256(nd * sizeof(float));
    const size_t n_b  = align256((size_t)N * sizeof(float));

    char* ws = (char*)d_ws;
    float* bufA   = (float*)ws;                       // h scratch
    float* bufB   = (float*)(ws + nd_b);              // z scratch
    float* deg    = (float*)(ws + 2 * nd_b);
    float* dinv   = (float*)(ws + 2 * nd_b + n_b);
    float* gsum   = (float*)(ws + 2 * nd_b + 2 * n_b);   // 64 floats
    float* gsumsq = gsum + FD;                            // 64 floats (contig)
    float* outf   = (float*)d_out;

    // Degrees (with self-loop) -> dinv
    k_init_deg <<<(N + 255) / 256, 256, 0, stream>>>(deg, N);
    k_count_deg<<<(E + 255) / 256, 256, 0, stream>>>(dst, deg, E);
    k_make_dinv<<<(N + 255) / 256, 256, 0, stream>>>(deg, dinv, N);

    const int nd_blocks = (int)((nd + 255) / 256);
    const float* in = x;

    for (int l = 0; l < NLAYERS; ++l) {
        float* h = bufA;
        float* z = (l == 1) ? outf : bufB;   // layer1's z doubles as layer2 input

        k_zero<<<nd_blocks, 256, 0, stream>>>(z, nd);
        k_zero<<<1, 128, 0, stream>>>(gsum, 2 * FD);

        gcn_gemm_wmma<<<N / GROWS, 128, 0, stream>>>(in, Ws + (size_t)l * FD * FD, h, N);
        k_edge_agg<<<(E + 15) / 16, 256, 0, stream>>>(h, src, dst, dinv, z, E);
        k_self_bias_stats<<<(N + 63) / 64, 256, 0, stream>>>(z, h, dinv, bs + l * FD,
                                                             gsum, gsumsq, N);
        if (l < NLAYERS - 1) {
            k_bn_norm<<<nd_blocks, 256, 0, stream>>>(z, z, gsum, gsumsq,
                                                     gammas + l * FD, betas + l * FD,
                                                     N, 1);
            in = z;
        } else {
            k_bn_norm<<<nd_blocks, 256, 0, stream>>>(z, outf, gsum, gsumsq,
                                                     gammas + l * FD, betas + l * FD,
                                                     N, 0);
        }
    }
}